// WaveDynamicsLayer_19207093747867
// MI455X (gfx1250) — compile-verified
//
#include <hip/hip_runtime.h>
#include <math.h>

// ---------------------------------------------------------------------------
// WaveDynamicsLayer for MI455X (gfx1250, wave32)
//   proj = (x @ [Wre0|Wim0|Wre1|Wim1] + bias) * (1-r_i)   (WMMA f32 GEMM)
//   chunked constant-coefficient complex scan
//   out  = sigmoid(h @ Wgate + bgate) * h                 (WMMA f32 GEMM)
// GEMMs: double-buffered LDS tiles filled with GLOBAL_LOAD_ASYNC_TO_LDS
// (ASYNCcnt), fragment-ready B layout -> ds_load_b64 straight into WMMA
// operand pairs, one barrier per K-chunk. Double-buffer selection is done
// with integer offsets into a single __shared__ object so InferAddressSpaces
// keeps the tile reads on the DS path (not FLAT).
// ---------------------------------------------------------------------------

typedef __attribute__((ext_vector_type(2))) float v2f;
typedef __attribute__((ext_vector_type(8))) float v8f;

#define WMMA_F32(a, b, c) \
    __builtin_amdgcn_wmma_f32_16x16x4_f32(false, (a), false, (b), (short)0, (c), false, false)

// Problem constants (match reference)
#define BB      4
#define SS      4096
#define KDIM    1024         // D_IN == D_MODEL
#define DMODEL  1024
#define DPER    512
#define LDPROJ  2048         // proj row stride: [re0|im0|re1|im1]
#define NCHUNK  32
#define CHUNK   128          // SS / NCHUNK
#define NCH     4096         // B * NSCALES * DPER independent recurrences
#define NKCH    (KDIM / 16)  // 64 K-chunks of 16

// LDS tile geometry
#define AROW 20              // A row stride (floats): 16B-aligned, conflict-free
#define BROW 80              // B k-pair row stride (float2): halves 32 banks apart
#define ASZ  (128 * AROW)    // floats per A buffer
#define BSZ  (8 * BROW * 2)  // floats per B buffer

// ---- CDNA5 async HBM->LDS copies (ASYNCcnt) -------------------------------
__device__ __forceinline__ void async_b128(const float* g, float* l) {
    unsigned lo = (unsigned)(uintptr_t)l;                 // generic->LDS = low 32b
    unsigned long long ga = (unsigned long long)(uintptr_t)g;
    asm volatile("global_load_async_to_lds_b128 %0, %1, off"
                 :: "v"(lo), "v"(ga) : "memory");
}
__device__ __forceinline__ void async_b32(const float* g, float* l) {
    unsigned lo = (unsigned)(uintptr_t)l;
    unsigned long long ga = (unsigned long long)(uintptr_t)g;
    asm volatile("global_load_async_to_lds_b32 %0, %1, off"
                 :: "v"(lo), "v"(ga) : "memory");
}
__device__ __forceinline__ void wait_async0() {
    asm volatile("s_wait_asynccnt 0x0" ::: "memory");
}

// r/theta per scale: linspace(R_MAX,R_MIN,2) / linspace(TH_MIN,TH_MAX,2)
__device__ __forceinline__ void scale_params(int scale, float& rv, float& th) {
    rv = scale ? 0.9f : 0.999f;
    th = scale ? 1.0f : 0.01f;
}

// ---------------------------------------------------------------------------
// Shared GEMM core: C(128x64) += A(128xK) @ B(KxLDB, cols colBase..+63).
// 256 threads = 8 waves, each wave owns a 32x32 register tile (4 WMMA accs).
// Pipeline per 16-wide K-chunk:
//   s_wait_asynccnt 0 ; barrier ; issue async(kc+1) ; 16x WMMA from LDS
// ---------------------------------------------------------------------------
template<int LDB>
__device__ __forceinline__ void wmma_gemm_core(
    const float* __restrict__ A, const float* __restrict__ B,
    int rowBase, int colBase,
    float (&As)[2 * ASZ], float (&Bs)[2 * BSZ], v8f acc[2][2])
{
    const int tid   = threadIdx.x;
    const int lane  = tid & 31;
    const int wave  = tid >> 5;
    const int col16 = lane & 15;
    const int half  = lane >> 4;
    const int waveM = wave & 3;      // *32 -> M offset in block
    const int waveN = wave >> 2;     // *32 -> N offset in block

    const int aRow = tid >> 2;       // 0..63 (+64 second pass)
    const int aK4  = (tid & 3) * 4;  // 0,4,8,12
    const int bK2  = wave;           // k-pair row handled by this wave (0..7)
    const int bN   = lane * 2;       // 0..62

    const float* srcA = A + (size_t)(rowBase + aRow) * KDIM + aK4;
    const float* srcB = B + (size_t)(2 * bK2) * LDB + colBase + bN;

    // Per-thread LDS destination offsets (within one buffer)
    const int dstA0 = aRow * AROW + aK4;
    const int dstA1 = (aRow + 64) * AROW + aK4;
    const int dstB  = (bK2 * BROW + bN) * 2;

    auto issue = [&](int kc, int buf) {
        const int ab = buf ? ASZ : 0;
        const int bb = buf ? BSZ : 0;
        // A tile 128x16: contiguous 16B per thread, 2 rows
        const float* a0 = srcA + kc * 16;
        async_b128(a0,                     &As[ab + dstA0]);
        async_b128(a0 + (size_t)64 * KDIM, &As[ab + dstA1]);
        // B tile 16x64 -> fragment-ready float2{B[k][n], B[k+1][n]}
        const float* b0 = srcB + (size_t)kc * 16 * LDB;
        async_b32(b0,           &Bs[bb + dstB + 0]);   // x of n
        async_b32(b0 + 1,       &Bs[bb + dstB + 2]);   // x of n+1
        async_b32(b0 + LDB,     &Bs[bb + dstB + 1]);   // y of n
        async_b32(b0 + LDB + 1, &Bs[bb + dstB + 3]);   // y of n+1
    };

    // Per-thread LDS read offsets (within one buffer)
    const int rdA = (waveM * 32 + col16) * AROW + 2 * half;   // + kw
    const int rdB = half * BROW * 2 + (waveN * 32 + col16) * 2; // + (kw>>1)*BROW*2

    issue(0, 0);
    for (int kc = 0; kc < NKCH; ++kc) {
        const int buf = kc & 1;
        wait_async0();           // this wave's chunk-kc DMA complete
        __syncthreads();         // all waves: chunk kc in LDS, chunk kc-1 reads done
        if (kc + 1 < NKCH) issue(kc + 1, buf ^ 1);  // overlaps with WMMAs below

        const int ab = buf ? ASZ : 0;
        const int bb = buf ? BSZ : 0;
        #pragma unroll
        for (int kw = 0; kw < 16; kw += 4) {
            v2f a0, a1, b0, b1;
            const int ia = ab + rdA + kw;                 // K pair base (even)
            a0.x = As[ia];              a0.y = As[ia + 1];
            a1.x = As[ia + 16 * AROW];  a1.y = As[ia + 16 * AROW + 1];
            const int ib = bb + rdB + (kw >> 1) * BROW * 2;
            b0.x = Bs[ib];      b0.y = Bs[ib + 1];        // one aligned b64
            b1.x = Bs[ib + 32]; b1.y = Bs[ib + 33];       // +16 columns
            acc[0][0] = WMMA_F32(a0, b0, acc[0][0]);
            acc[0][1] = WMMA_F32(a0, b1, acc[0][1]);
            acc[1][0] = WMMA_F32(a1, b0, acc[1][0]);
            acc[1][1] = WMMA_F32(a1, b1, acc[1][1]);
        }
    }
}

// ---------------------------------------------------------------------------
// GEMM 1: proj[16384][2048] = (x @ Wcat + biascat) * (1 - r_i)
// ---------------------------------------------------------------------------
__global__ __launch_bounds__(256) void gemm_proj_kernel(
    const float* __restrict__ X,
    const float* __restrict__ Wre0, const float* __restrict__ bre0,
    const float* __restrict__ Wim0, const float* __restrict__ bim0,
    const float* __restrict__ Wre1, const float* __restrict__ bre1,
    const float* __restrict__ Wim1, const float* __restrict__ bim1,
    float* __restrict__ proj)
{
    __shared__ float As[2 * ASZ];
    __shared__ float Bs[2 * BSZ];

    const int lane  = threadIdx.x & 31;
    const int wave  = threadIdx.x >> 5;
    const int col16 = lane & 15;
    const int half  = lane >> 4;
    const int waveM = wave & 3;
    const int waveN = wave >> 2;
    const int rowBase = blockIdx.y * 128;
    const int colBase = blockIdx.x * 64;

    // 64-wide N block never straddles a 512 boundary -> fixed source array.
    const int which = colBase >> 9;
    const float* Wsel = (which == 0) ? Wre0 : (which == 1) ? Wim0 : (which == 2) ? Wre1 : Wim1;
    const float* bsel = (which == 0) ? bre0 : (which == 1) ? bim0 : (which == 2) ? bre1 : bim1;
    const float scaleF = (colBase >> 10) ? (1.0f - 0.9f) : (1.0f - 0.999f);

    v8f acc[2][2] = {};
    wmma_gemm_core<DPER>(X, Wsel, rowBase, colBase & 511, As, Bs, acc);

    #pragma unroll
    for (int i = 0; i < 2; ++i) {
        #pragma unroll
        for (int j = 0; j < 2; ++j) {
            const int gN = colBase + waveN * 32 + j * 16 + col16;
            const float bias = bsel[gN & 511];
            #pragma unroll
            for (int r = 0; r < 8; ++r) {
                const int gM = rowBase + waveM * 32 + i * 16 + r + 8 * half;
                proj[(size_t)gM * LDPROJ + gN] = (acc[i][j][r] + bias) * scaleF;
            }
        }
    }
}

// ---------------------------------------------------------------------------
// Scan pass 1: chunk-local complex scan, in place over proj; final state out.
// tid -> d(9b) | scale(1b) | b(2b) | chunk(5b); lanes = consecutive d.
// ---------------------------------------------------------------------------
__global__ __launch_bounds__(256) void scan_pass1_kernel(
    float* __restrict__ proj, float2* __restrict__ states)
{
    const int tid   = blockIdx.x * 256 + threadIdx.x;  // 0 .. 131071
    const int d     = tid & 511;
    const int scale = (tid >> 9) & 1;
    const int b     = (tid >> 10) & 3;
    const int c     = tid >> 12;
    const int ch    = tid & 4095;

    float rv, th; scale_params(scale, rv, th);
    const float aR = rv * cosf(th);
    const float aI = rv * sinf(th);

    size_t idx = ((size_t)(b * SS + c * CHUNK)) * LDPROJ + scale * 1024 + d;
    float hr = 0.0f, hi = 0.0f;
    for (int t = 0; t < CHUNK; ++t) {
        const float br = proj[idx];
        const float bi = proj[idx + 512];
        const float nr = fmaf(aR, hr, fmaf(-aI, hi, br));
        const float ni = fmaf(aR, hi, fmaf( aI, hr, bi));
        hr = nr; hi = ni;
        proj[idx]       = hr;     // overwrite b with local h (read-before-write)
        proj[idx + 512] = hi;
        idx += LDPROJ;
    }
    states[c * NCH + ch] = make_float2(hr, hi);
}

// ---------------------------------------------------------------------------
// Scan pass 2+3 fused: combine previous chunk states with a^CHUNK, then fix
// up: h_t = local_h_t + a^(k+1) * h_in; only the real part is emitted.
// ---------------------------------------------------------------------------
__global__ __launch_bounds__(256) void scan_fix_kernel(
    const float* __restrict__ proj, const float2* __restrict__ states,
    float* __restrict__ hout)
{
    const int tid   = blockIdx.x * 256 + threadIdx.x;
    const int d     = tid & 511;
    const int scale = (tid >> 9) & 1;
    const int b     = (tid >> 10) & 3;
    const int c     = tid >> 12;
    const int ch    = tid & 4095;

    float rv, th; scale_params(scale, rv, th);
    const float aR = rv * cosf(th);
    const float aI = rv * sinf(th);

    // a^CHUNK
    const float rL  = powf(rv, (float)CHUNK);
    const float aLR = rL * cosf(th * (float)CHUNK);
    const float aLI = rL * sinf(th * (float)CHUNK);

    float hinR = 0.0f, hinI = 0.0f;
    for (int j = 0; j < c; ++j) {
        const float2 s = states[j * NCH + ch];
        const float nr = fmaf(aLR, hinR, fmaf(-aLI, hinI, s.x));
        const float ni = fmaf(aLR, hinI, fmaf( aLI, hinR, s.y));
        hinR = nr; hinI = ni;
    }

    float pR = aR, pI = aI;   // a^(k+1), k = 0 at chunk start
    size_t idx  = ((size_t)(b * SS + c * CHUNK)) * LDPROJ + scale * 1024 + d;
    size_t oidx = ((size_t)(b * SS + c * CHUNK)) * DMODEL + scale * DPER + d;
    for (int t = 0; t < CHUNK; ++t) {
        const float hlr = proj[idx];                       // local real only
        hout[oidx] = hlr + pR * hinR - pI * hinI;
        const float nR = pR * aR - pI * aI;
        const float nI = pR * aI + pI * aR;
        pR = nR; pI = nI;
        idx  += LDPROJ;
        oidx += DMODEL;
    }
}

// ---------------------------------------------------------------------------
// GEMM 2: out = sigmoid(h @ Wgate + bgate) * h
// ---------------------------------------------------------------------------
__global__ __launch_bounds__(256) void gemm_gate_kernel(
    const float* __restrict__ H, const float* __restrict__ Wg,
    const float* __restrict__ bg, float* __restrict__ out)
{
    __shared__ float As[2 * ASZ];
    __shared__ float Bs[2 * BSZ];

    const int lane  = threadIdx.x & 31;
    const int wave  = threadIdx.x >> 5;
    const int col16 = lane & 15;
    const int half  = lane >> 4;
    const int waveM = wave & 3;
    const int waveN = wave >> 2;
    const int rowBase = blockIdx.y * 128;
    const int colBase = blockIdx.x * 64;

    v8f acc[2][2] = {};
    wmma_gemm_core<DMODEL>(H, Wg, rowBase, colBase, As, Bs, acc);

    #pragma unroll
    for (int i = 0; i < 2; ++i) {
        #pragma unroll
        for (int j = 0; j < 2; ++j) {
            const int gN = colBase + waveN * 32 + j * 16 + col16;
            const float bias = bg[gN];
            #pragma unroll
            for (int r = 0; r < 8; ++r) {
                const int gM = rowBase + waveM * 32 + i * 16 + r + 8 * half;
                const size_t off = (size_t)gM * DMODEL + gN;
                const float z = acc[i][j][r] + bias;
                const float g = 1.0f / (1.0f + expf(-z));
                out[off] = g * H[off];
            }
        }
    }
}

// ---------------------------------------------------------------------------
// Host-side launcher
// ---------------------------------------------------------------------------
extern "C" void kernel_launch(void* const* d_in, const int* in_sizes, int n_in,
                              void* d_out, int out_size, void* d_ws, size_t ws_size,
                              hipStream_t stream) {
    const float* x     = (const float*)d_in[0];
    const float* Wre0  = (const float*)d_in[1];
    const float* bre0  = (const float*)d_in[2];
    const float* Wim0  = (const float*)d_in[3];
    const float* bim0  = (const float*)d_in[4];
    const float* Wre1  = (const float*)d_in[5];
    const float* bre1  = (const float*)d_in[6];
    const float* Wim1  = (const float*)d_in[7];
    const float* bim1  = (const float*)d_in[8];
    const float* Wgate = (const float*)d_in[9];
    const float* bgate = (const float*)d_in[10];
    float* out = (float*)d_out;

    // Workspace layout
    const size_t projBytes   = (size_t)BB * SS * LDPROJ * sizeof(float);   // 134.2 MB
    const size_t statesBytes = (size_t)NCHUNK * NCH * sizeof(float2);      //   1.0 MB
    float*  proj   = (float*)d_ws;
    float2* states = (float2*)((char*)d_ws + projBytes);
    float*  hbuf   = (float*)((char*)d_ws + projBytes + statesBytes);      //  67.1 MB

    gemm_proj_kernel<<<dim3(LDPROJ / 64, (BB * SS) / 128), 256, 0, stream>>>(
        x, Wre0, bre0, Wim0, bim0, Wre1, bre1, Wim1, bim1, proj);

    scan_pass1_kernel<<<(NCH * NCHUNK) / 256, 256, 0, stream>>>(proj, states);

    scan_fix_kernel<<<(NCH * NCHUNK) / 256, 256, 0, stream>>>(proj, states, hbuf);

    gemm_gate_kernel<<<dim3(DMODEL / 64, (BB * SS) / 128), 256, 0, stream>>>(
        hbuf, Wgate, bgate, out);
}